// GraphFeature_28956669509832
// MI455X (gfx1250) — compile-verified
//
#include <hip/hip_runtime.h>
#include <hip/hip_bf16.h>

typedef __attribute__((ext_vector_type(2))) float v2f;
typedef __attribute__((ext_vector_type(8))) float v8f;
typedef __attribute__((ext_vector_type(4))) int   v4i;

#define NPTS 4096
#define CH   16
#define TOPK 20
#define MASK_FILL -1.0e9f
#define NEG_BIG  -3.0e38f

// ---------------------------------------------------------------------------
// Kernel 1: xx[b,m] = sum_c x[b,c,m]^2   (8*4096 floats into workspace)
// ---------------------------------------------------------------------------
__global__ void gf_xx_kernel(const float* __restrict__ x, float* __restrict__ xx) {
    int t = blockIdx.x * blockDim.x + threadIdx.x;      // 0 .. 32767
    int b = t >> 12;
    int m = t & (NPTS - 1);
    const float* xp = x + (size_t)b * CH * NPTS + m;
    float s = 0.0f;
#pragma unroll
    for (int c = 0; c < CH; ++c) {
        float v = xp[c * NPTS];
        s += v * v;
    }
    xx[t] = s;
}

// ---------------------------------------------------------------------------
// Kernel 2: fused  WMMA-gram -> pd -> mask -> top-20 -> gather/concat output
// Block = 256 threads (8 wave32). Block handles (batch bb, rows n0..n0+15).
// ---------------------------------------------------------------------------
__global__ void __launch_bounds__(256)
gf_fused_kernel(const float* __restrict__ x, const int* __restrict__ mask,
                const float* __restrict__ xx, float* __restrict__ out) {
    __shared__ float pdT[8][256];          // per-wave 16x16 pd tile (unmasked)
    __shared__ float candV[256 * TOPK];    // per-lane sorted top-20 values
    __shared__ int   candI[256 * TOPK];    // per-lane sorted top-20 indices
    __shared__ float topV[16][TOPK];
    __shared__ int   topI[16][TOPK];

    const int tid  = threadIdx.x;
    const int wave = tid >> 5;
    const int lane = tid & 31;
    const int bb   = blockIdx.x >> 8;            // batch 0..7
    const int n0   = (blockIdx.x & 255) << 4;    // row tile base

    const int hh    = lane >> 4;                 // lane half (0/1)
    const int lcol  = lane & 15;                 // tile column for B/C frags
    const int kbase = hh * 2;                    // A/B frag K base per lane half

    const float* xb = x + (size_t)bb * CH * NPTS;

    // ---- A fragment: rows n0..n0+15, K=16 as four 16x16x4 chunks ----------
    // f32 A 16x4 layout: lane L holds M = L%16; VGPR0 -> K = 2*(L/16),
    // VGPR1 -> K = 2*(L/16)+1.
    v2f afrag[4];
#pragma unroll
    for (int j = 0; j < 4; ++j) {
        afrag[j].x = xb[(4 * j + kbase)     * NPTS + n0 + lcol];
        afrag[j].y = xb[(4 * j + kbase + 1) * NPTS + n0 + lcol];
    }

    // xx for the 8 rows this lane's C VGPRs map to (M = v + hh*8)
    float xxn[8];
#pragma unroll
    for (int v = 0; v < 8; ++v)
        xxn[v] = xx[bb * NPTS + n0 + v + hh * 8];

    // ---- register-resident sorted top-20 (descending) ---------------------
    float tv[TOPK];
    int   ti[TOPK];
#pragma unroll
    for (int q = 0; q < TOPK; ++q) { tv[q] = NEG_BIG; ti[q] = -1; }

    const int row_r = lane >> 1;        // row this lane scans (0..15)
    const int cofs  = (lane & 1) * 8;   // column sub-range (0 or 8)

    // loop-invariant mask base for this lane's half-row (8 consecutive ints)
    const int* mrow = mask + ((size_t)bb * NPTS + n0 + row_r) * NPTS + cofs;

    // ---- main loop over column tiles (32 iterations per wave, uniform) ----
    for (int mt = wave; mt < NPTS / 16; mt += 8) {
        const int m0 = mt << 4;

        // B fragment: same striping as A, columns m0..m0+15
        v2f bfrag[4];
#pragma unroll
        for (int j = 0; j < 4; ++j) {
            bfrag[j].x = xb[(4 * j + kbase)     * NPTS + m0 + lcol];
            bfrag[j].y = xb[(4 * j + kbase + 1) * NPTS + m0 + lcol];
        }

        v8f c = {};
        c = __builtin_amdgcn_wmma_f32_16x16x4_f32(false, afrag[0], false, bfrag[0],
                                                  (short)0, c, false, false);
        c = __builtin_amdgcn_wmma_f32_16x16x4_f32(false, afrag[1], false, bfrag[1],
                                                  (short)0, c, false, false);
        c = __builtin_amdgcn_wmma_f32_16x16x4_f32(false, afrag[2], false, bfrag[2],
                                                  (short)0, c, false, false);
        c = __builtin_amdgcn_wmma_f32_16x16x4_f32(false, afrag[3], false, bfrag[3],
                                                  (short)0, c, false, false);

        // mask for this lane's half-row: two NT b128 loads (streaming 512 MB)
        v4i mk0 = __builtin_nontemporal_load((const v4i*)(mrow + m0));
        v4i mk1 = __builtin_nontemporal_load((const v4i*)(mrow + m0 + 4));

        const float xxm = xx[bb * NPTS + m0 + lcol];

        // epilogue: pd = 2*inner - xx[n] - xx[m]; stage to LDS (unmasked)
#pragma unroll
        for (int v = 0; v < 8; ++v) {
            float pd = 2.0f * c[v] - xxn[v] - xxm;
            pdT[wave][(v + hh * 8) * 16 + lcol] = pd;
        }
        __syncthreads();   // uniform: every wave runs exactly 32 iterations

        // each lane scans one half-row, applies mask, inserts into its top-20
#pragma unroll
        for (int q = 0; q < 8; ++q) {
            float pd   = pdT[wave][row_r * 16 + cofs + q];
            int   mk   = (q < 4) ? mk0[q] : mk1[q - 4];
            float val  = (mk == 1) ? pd : MASK_FILL;
            int   gidx = m0 + cofs + q;
            if (val > tv[TOPK - 1]) {
                tv[TOPK - 1] = val;
                ti[TOPK - 1] = gidx;
#pragma unroll
                for (int p = TOPK - 1; p > 0; --p) {
                    if (tv[p] > tv[p - 1]) {
                        float tf = tv[p]; tv[p] = tv[p - 1]; tv[p - 1] = tf;
                        int   tn = ti[p]; ti[p] = ti[p - 1]; ti[p - 1] = tn;
                    }
                }
            }
        }
        __syncthreads();
    }

    // ---- dump per-lane lists, merge 16 sorted lists per row ---------------
#pragma unroll
    for (int q = 0; q < TOPK; ++q) {
        candV[tid * TOPK + q] = tv[q];
        candI[tid * TOPK + q] = ti[q];
    }
    __syncthreads();

    if (tid < 16) {
        const int row = tid;
        for (int j = 0; j < TOPK; ++j) {
            float best = NEG_BIG; int bi = -1, bs = -1;
            for (int w = 0; w < 8; ++w) {
                for (int e = 0; e < 2; ++e) {
                    const int base = (w * 32 + row * 2 + e) * TOPK;
                    for (int q = 0; q < TOPK; ++q) {
                        float v = candV[base + q];
                        if (v > best) { best = v; bi = candI[base + q]; bs = base + q; }
                    }
                }
            }
            topV[row][j] = best;
            topI[row][j] = bi;
            if (bs >= 0) candV[bs] = NEG_BIG;
        }
    }
    __syncthreads();

    // ---- gather + concat output: out[b, 2C, n, k] -------------------------
    for (int pair = tid; pair < 16 * TOPK; pair += 256) {
        const int row = pair / TOPK;
        const int j   = pair - row * TOPK;
        const int n   = n0 + row;
        int gi = topI[row][j];
        if (gi < 0) gi = 0;
#pragma unroll
        for (int ch = 0; ch < CH; ++ch) {
            float f   = xb[ch * NPTS + gi];
            float cen = xb[ch * NPTS + n];
            __builtin_nontemporal_store(
                f - cen, &out[(((size_t)bb * 2 * CH + ch) * NPTS + n) * TOPK + j]);
            __builtin_nontemporal_store(
                cen, &out[(((size_t)bb * 2 * CH + ch + CH) * NPTS + n) * TOPK + j]);
        }
    }
}

// ---------------------------------------------------------------------------
extern "C" void kernel_launch(void* const* d_in, const int* in_sizes, int n_in,
                              void* d_out, int out_size, void* d_ws, size_t ws_size,
                              hipStream_t stream) {
    const float* x    = (const float*)d_in[0];   // [8,16,4096] f32
    const int*   mask = (const int*)d_in[1];     // [8,4096,4096] i32
    float* xx  = (float*)d_ws;                   // 8*4096 f32 scratch (128 KB)
    float* out = (float*)d_out;                  // [8,32,4096,20] f32

    gf_xx_kernel<<<(8 * NPTS) / 256, 256, 0, stream>>>(x, xx);
    gf_fused_kernel<<<8 * (NPTS / 16), 256, 0, stream>>>(x, mask, xx, out);
}